// PulseFieldWrapper_73967926772320
// MI455X (gfx1250) — compile-verified
//
#include <hip/hip_runtime.h>
#include <math.h>

#define VOCAB 50257
#define DIM   1024
#define BATCH 8
#define SEQ   4096
#define NCHUNK 64
#define CHUNK (SEQ / NCHUNK)   // 64 = 2^6
#define LOG2_CHUNK 6
#define APAD  8
#define ALD   (DIM + APAD)     // 1032 floats per padded LDS row (bank-conflict-free)

typedef __attribute__((ext_vector_type(2))) float v2f;
typedef __attribute__((ext_vector_type(8))) float v8f;
typedef int v4i __attribute__((vector_size(16)));

__device__ __forceinline__ float sigmoidf_(float x) {
    return 1.0f / (1.0f + __expf(-x));
}

// ---------------------------------------------------------------------------
// Phase 1a: per-chunk local EMA (h0=0). grid = (NCHUNK, BATCH), block = DIM.
// Token index is block-uniform per step -> scalar load; emb row coalesced.
// ---------------------------------------------------------------------------
__global__ void __launch_bounds__(DIM)
ema_partial_kernel(const int* __restrict__ ids,
                   const float* __restrict__ emb,
                   const float* __restrict__ enc_decay,
                   float* __restrict__ part)   // (BATCH, NCHUNK, DIM)
{
    const int d = threadIdx.x;
    const int k = blockIdx.x;
    const int b = blockIdx.y;
    const float a = sigmoidf_(enc_decay[d]);
    const float om = 1.0f - a;
    const int* idp = ids + b * SEQ + k * CHUNK;
    float c = 0.0f;
#pragma unroll 8
    for (int t = 0; t < CHUNK; ++t) {
        const int tok = idp[t];
        const float x = emb[(size_t)tok * DIM + d];
        c = fmaf(a, c, om * x);
    }
    part[((size_t)(b * NCHUNK + k)) * DIM + d] = c;
}

// ---------------------------------------------------------------------------
// Phase 1b: exact chunk combine via Horner with a^CHUNK.
// ---------------------------------------------------------------------------
__global__ void __launch_bounds__(DIM)
ema_combine_kernel(const float* __restrict__ part,
                   const float* __restrict__ enc_decay,
                   float* __restrict__ h_out)  // (BATCH, DIM)
{
    const int d = threadIdx.x;
    const int b = blockIdx.x;
    float a = sigmoidf_(enc_decay[d]);
    float aC = a;
#pragma unroll
    for (int i = 0; i < LOG2_CHUNK; ++i) aC *= aC;   // a^CHUNK
    float h = 0.0f;
#pragma unroll 8
    for (int k = 0; k < NCHUNK; ++k)
        h = fmaf(aC, h, part[((size_t)(b * NCHUNK + k)) * DIM + d]);
    h_out[b * DIM + d] = h;
}

// ---------------------------------------------------------------------------
// WMMA mainloop: C(16 x 16-tile) = A(16xDIM, rows 8..15 zero) * W^T.
// A in padded LDS; W row-major (N x DIM), streamed with NT hint (read-once).
// ---------------------------------------------------------------------------
__device__ __forceinline__ v8f wmma_row_gemm(const float* As,
                                             const float* __restrict__ W,
                                             int nClamped, v8f c)
{
    const int lane = threadIdx.x & 31;
    const int m    = lane & 15;
    const int koff = (lane >> 4) << 1;              // 0 (lanes 0-15) or 2
    const float* aptr = As + m * ALD + koff;        // rows 8..15 are zero
    const float* wptr = W + (size_t)nClamped * DIM + koff;
#pragma unroll 8
    for (int k0 = 0; k0 < DIM; k0 += 4) {
        const v2f afrag = *(const v2f*)(aptr + k0);                      // ds_load
        const v2f bfrag = __builtin_nontemporal_load((const v2f*)(wptr + k0)); // NT stream
        c = __builtin_amdgcn_wmma_f32_16x16x4_f32(
                /*neg_a=*/false, afrag, /*neg_b=*/false, bfrag,
                /*c_mod=*/(short)0, c, /*reuse_a=*/false, /*reuse_b=*/false);
    }
    return c;
}

// ---------------------------------------------------------------------------
// Stage an (8 x DIM) activation into a padded 16-row LDS tile, rows 8..15 = 0.
// Copy-only variant uses the CDNA5 async global->LDS path when available.
// ---------------------------------------------------------------------------
__device__ __forceinline__ void stage_A_tanh(float* As, const float* __restrict__ src)
{
    for (int i = threadIdx.x; i < 16 * DIM; i += blockDim.x) {
        const int r = i >> 10, cc = i & (DIM - 1);
        As[r * ALD + cc] = (r < 8) ? tanhf(src[r * DIM + cc]) : 0.0f;
    }
    __syncthreads();
}

__device__ __forceinline__ void stage_A_copy(float* As, const float* __restrict__ src)
{
#if __has_builtin(__builtin_amdgcn_global_load_async_to_lds_b128)
    // rows 0..7: GLOBAL_LOAD_ASYNC_TO_LDS_B128 (ASYNCcnt-tracked DMA-style copy)
    for (int j = threadIdx.x; j < 8 * (DIM / 4); j += blockDim.x) {
        const int r  = j >> 8;                 // 256 float4 per row
        const int cc = (j & 255) << 2;
        const v4i* gsrc = (const v4i*)(src + r * DIM + cc);
        v4i*       ldst = (v4i*)(As + r * ALD + cc);
        __builtin_amdgcn_global_load_async_to_lds_b128(
            (__attribute__((address_space(1))) v4i*)gsrc,
            (__attribute__((address_space(3))) v4i*)ldst,
            0, 0);
    }
    // rows 8..15: zero-fill with plain LDS stores
    for (int j = threadIdx.x; j < 8 * (DIM / 4); j += blockDim.x) {
        const int r  = 8 + (j >> 8);
        const int cc = (j & 255) << 2;
        float4 z = make_float4(0.f, 0.f, 0.f, 0.f);
        *(float4*)(As + r * ALD + cc) = z;
    }
#if __has_builtin(__builtin_amdgcn_s_wait_asynccnt)
    __builtin_amdgcn_s_wait_asynccnt(0);
#else
    asm volatile("s_wait_asynccnt 0x0" ::: "memory");
#endif
    __syncthreads();
#else
    for (int i = threadIdx.x; i < 16 * DIM; i += blockDim.x) {
        const int r = i >> 10, cc = i & (DIM - 1);
        As[r * ALD + cc] = (r < 8) ? src[r * DIM + cc] : 0.0f;
    }
    __syncthreads();
#endif
}

// ---------------------------------------------------------------------------
// Phase 2a: V = tanh(h); gate = sigmoid(V @ gate_W^T + gate_b);
//           Hc = sigmoid(cry_decay)*h + gate*V.    64 tiles -> 8 blocks.
// ---------------------------------------------------------------------------
__global__ void __launch_bounds__(256)
gate_crystal_kernel(const float* __restrict__ h,
                    const float* __restrict__ gate_W,
                    const float* __restrict__ gate_b,
                    const float* __restrict__ cry_decay,
                    float* __restrict__ Hc)
{
    __shared__ float As[16 * ALD];
    stage_A_tanh(As, h);                             // As = tanh(h) = V
    const int wave = (blockIdx.x * blockDim.x + threadIdx.x) >> 5;
    const int lane = threadIdx.x & 31;
    const int n = wave * 16 + (lane & 15);
    v8f c = {};
    c = wmma_row_gemm(As, gate_W, n, c);
    if (lane < 16) {
        const float gb  = gate_b[n];
        const float cdj = sigmoidf_(cry_decay[n]);
#pragma unroll
        for (int r = 0; r < 8; ++r) {
            const float g  = sigmoidf_(c[r] + gb);
            const float Vv = As[r * ALD + n];
            const float hv = h[r * DIM + n];
            Hc[r * DIM + n] = fmaf(cdj, hv, g * Vv);
        }
    }
}

// ---------------------------------------------------------------------------
// Phase 2b: Vp = tanh(Hc @ cry_W^T + cry_b)
// ---------------------------------------------------------------------------
__global__ void __launch_bounds__(256)
crystal_dense_kernel(const float* __restrict__ Hc,
                     const float* __restrict__ cry_W,
                     const float* __restrict__ cry_b,
                     float* __restrict__ Vp)
{
    __shared__ float As[16 * ALD];
    stage_A_copy(As, Hc);
    const int wave = (blockIdx.x * blockDim.x + threadIdx.x) >> 5;
    const int lane = threadIdx.x & 31;
    const int n = wave * 16 + (lane & 15);
    v8f c = {};
    c = wmma_row_gemm(As, cry_W, n, c);
    if (lane < 16) {
        const float bb = cry_b[n];
#pragma unroll
        for (int r = 0; r < 8; ++r)
            Vp[r * DIM + n] = tanhf(c[r] + bb);
    }
}

// ---------------------------------------------------------------------------
// Phase 3: logits = Vp @ dec_W^T + dec_b.  3142 N-tiles, 8 waves/block.
// Streams 206 MB of dec_W exactly once (NT) — the bandwidth-critical kernel.
// ---------------------------------------------------------------------------
__global__ void __launch_bounds__(256)
decoder_kernel(const float* __restrict__ Vp,
               const float* __restrict__ dec_W,
               const float* __restrict__ dec_b,
               float* __restrict__ out)
{
    __shared__ float As[16 * ALD];
    stage_A_copy(As, Vp);
    const int wave = (blockIdx.x * blockDim.x + threadIdx.x) >> 5;
    const int n0 = wave * 16;
    if (n0 >= VOCAB) return;                         // wave-uniform exit
    const int lane = threadIdx.x & 31;
    const int n  = n0 + (lane & 15);
    const int nc = (n < VOCAB) ? n : (VOCAB - 1);    // clamp tail-tile loads
    const float bias = dec_b[nc];
    v8f c = {bias, bias, bias, bias, bias, bias, bias, bias};
    c = wmma_row_gemm(As, dec_W, nc, c);
    if (lane < 16 && n < VOCAB) {
#pragma unroll
        for (int r = 0; r < 8; ++r)
            out[(size_t)r * VOCAB + n] = c[r];
    }
}

// ---------------------------------------------------------------------------
extern "C" void kernel_launch(void* const* d_in, const int* in_sizes, int n_in,
                              void* d_out, int out_size, void* d_ws, size_t ws_size,
                              hipStream_t stream)
{
    const int*   ids       = (const int*)  d_in[0];
    const float* emb       = (const float*)d_in[1];
    const float* enc_decay = (const float*)d_in[2];
    const float* cry_decay = (const float*)d_in[3];
    const float* gate_W    = (const float*)d_in[4];
    const float* gate_b    = (const float*)d_in[5];
    const float* cry_W     = (const float*)d_in[6];
    const float* cry_b     = (const float*)d_in[7];
    const float* dec_W     = (const float*)d_in[8];
    const float* dec_b     = (const float*)d_in[9];
    float* out = (float*)d_out;

    float* part = (float*)d_ws;                          // B*NCHUNK*DIM  (2 MB)
    float* h    = part + (size_t)BATCH * NCHUNK * DIM;   // B*DIM
    float* Hc   = h  + BATCH * DIM;
    float* Vp   = Hc + BATCH * DIM;

    ema_partial_kernel<<<dim3(NCHUNK, BATCH), DIM, 0, stream>>>(ids, emb, enc_decay, part);
    ema_combine_kernel<<<BATCH, DIM, 0, stream>>>(part, enc_decay, h);
    gate_crystal_kernel <<<DIM / 16 / 8, 256, 0, stream>>>(h, gate_W, gate_b, cry_decay, Hc);
    crystal_dense_kernel<<<DIM / 16 / 8, 256, 0, stream>>>(Hc, cry_W, cry_b, Vp);

    const int tiles  = (VOCAB + 15) / 16;   // 3142
    const int blocks = (tiles + 7) / 8;     // 393
    decoder_kernel<<<blocks, 256, 0, stream>>>(Vp, dec_W, dec_b, out);
}